// MaskedLoss_15496242004751
// MI455X (gfx1250) — compile-verified
//
#include <hip/hip_runtime.h>

#define W_OUT  0.1f
#define EXPAND 1.2f
#define SH 160
#define NB 4

static const int N_ELEM = NB * SH * SH * SH;   // 16,384,000
static const int N4     = N_ELEM / 4;          // 4,096,000 float4 chunks
#define INV_N (1.0f / 16384000.0f)

typedef __attribute__((ext_vector_type(2))) float v2f;
typedef __attribute__((ext_vector_type(8))) float v8f;

// -------- workspace layout (ints) --------
// [0*NB*SH) anyD   [1*NB*SH) anyH   [2*NB*SH) anyW
// [3*NB*SH) insD   [4*NB*SH) insH   [5*NB*SH) insW
#define WS_INTS (6 * NB * SH)

// Kernel 1: zero the flag/inside tables and the scalar output (ws/out are poisoned).
__global__ void k_init(int* __restrict__ ws, float* __restrict__ out) {
    int i = blockIdx.x * blockDim.x + threadIdx.x;
    if (i < WS_INTS) ws[i] = 0;
    if (i == 0) out[0] = 0.0f;
}

// Kernel 2: per-(b,d) slice — OR-project mask>0 onto D, H, W axes.
// Same-value races (everyone stores 1) are benign; no atomics needed.
__global__ __launch_bounds__(256)
void k_project(const int* __restrict__ mask, int* __restrict__ anyD,
               int* __restrict__ anyH, int* __restrict__ anyW) {
    __shared__ int hF[SH];
    __shared__ int wF[SH];
    __shared__ int sAny;
    const int b = blockIdx.x / SH;
    const int d = blockIdx.x % SH;
    for (int i = threadIdx.x; i < SH; i += blockDim.x) { hF[i] = 0; wF[i] = 0; }
    if (threadIdx.x == 0) sAny = 0;
    __syncthreads();

    const int4* s4 = (const int4*)(mask + (size_t)(b * SH + d) * (SH * SH));
    for (int c = threadIdx.x; c < (SH * SH) / 4; c += blockDim.x) {
        int4 v = s4[c];                 // GLOBAL_LOAD_B128
        int base = c * 4;
        int h  = base / SH;
        int w0 = base - h * SH;         // 4 consecutive w in one row (160 % 4 == 0)
        if (v.x > 0) { hF[h] = 1; wF[w0 + 0] = 1; sAny = 1; }
        if (v.y > 0) { hF[h] = 1; wF[w0 + 1] = 1; sAny = 1; }
        if (v.z > 0) { hF[h] = 1; wF[w0 + 2] = 1; sAny = 1; }
        if (v.w > 0) { hF[h] = 1; wF[w0 + 3] = 1; sAny = 1; }
    }
    __syncthreads();
    if (threadIdx.x == 0 && sAny) anyD[b * SH + d] = 1;
    for (int i = threadIdx.x; i < SH; i += blockDim.x) {
        if (hF[i]) anyH[b * SH + i] = 1;
        if (wF[i]) anyW[b * SH + i] = 1;
    }
}

// Kernel 3: 12 tiny tasks (3 axes x 4 batches): bbox -> expanded interval -> inside table.
// Mirrors reference float semantics: c=(mx+mn)/2, e=(mx-mn+1)/2*EXPAND,
// lo=max(0,floor(c-e)), hi=min(SH-1,floor(c+e)), inside = [lo, hi) ; no fg -> all 0.
__global__ void k_bounds(int* __restrict__ ws) {
    const int t = threadIdx.x;
    if (t >= 3 * NB) return;
    const int axis = t / NB;
    const int b    = t % NB;
    const int* any = ws + axis * NB * SH + b * SH;
    int*       ins = ws + (3 + axis) * NB * SH + b * SH;
    int mn = -1, mx = -1;
    for (int i = 0; i < SH; ++i)
        if (any[i]) { if (mn < 0) mn = i; mx = i; }
    if (mn < 0) {
        for (int i = 0; i < SH; ++i) ins[i] = 0;   // no fg -> box false -> W_OUT everywhere
        return;
    }
    float c = (float)(mx + mn) * 0.5f;
    float e = ((float)(mx - mn) + 1.0f) * 0.5f * EXPAND;
    int lo_i = (int)fmaxf(0.0f, floorf(c - e));
    int hi_i = (int)fminf((float)(SH - 1), floorf(c + e));
    for (int i = 0; i < SH; ++i) ins[i] = (i >= lo_i && i < hi_i) ? 1 : 0;
}

// Kernel 4: streaming masked-MSE reduction.
// float4 loads (GLOBAL_LOAD_B128) + prefetch hints; inside tables hit near caches.
// Wave-level sum via V_WMMA_F32_16X16X4_F32: A = partials (zero-padded 16x4),
// B = ones(4x16) -> D[m][n] = partial(m) + partial(m+16); per-lane sum of the
// 8 D-VGPRs + one shfl_xor(16) gives the full 32-lane sum.
__global__ __launch_bounds__(256)
void k_mse(const float4* __restrict__ p4, const float4* __restrict__ t4,
           const int* __restrict__ ws, float* __restrict__ out) {
    const int* insD = ws + 3 * NB * SH;
    const int* insH = ws + 4 * NB * SH;
    const int* insW = ws + 5 * NB * SH;

    float acc = 0.0f;
    const int stride = gridDim.x * blockDim.x;
    for (int c = blockIdx.x * blockDim.x + threadIdx.x; c < N4; c += stride) {
        __builtin_prefetch(&p4[c + stride], 0, 0);   // global_prefetch_b8
        __builtin_prefetch(&t4[c + stride], 0, 0);
        float4 p = p4[c];
        float4 t = t4[c];
        int base = c * 4;
        int w0 = base % SH;
        int r  = base / SH;
        int h  = r % SH;
        int r2 = r / SH;
        int d  = r2 % SH;
        int b  = r2 / SH;
        int in_dh = insD[b * SH + d] & insH[b * SH + h];
        const int4 iw = *(const int4*)(insW + b * SH + w0);   // 16B-aligned (w0 % 4 == 0)
        float w, df;
        w = (in_dh & iw.x) ? 1.0f : W_OUT; df = w * (p.x - t.x); acc = fmaf(df, df, acc);
        w = (in_dh & iw.y) ? 1.0f : W_OUT; df = w * (p.y - t.y); acc = fmaf(df, df, acc);
        w = (in_dh & iw.z) ? 1.0f : W_OUT; df = w * (p.z - t.z); acc = fmaf(df, df, acc);
        w = (in_dh & iw.w) ? 1.0f : W_OUT; df = w * (p.w - t.w); acc = fmaf(df, df, acc);
    }

    // ---- wave32 reduction via WMMA (EXEC all-ones here: no divergence) ----
    v2f a;  a.x = acc;   a.y = 0.0f;     // lane L -> A[M=L%16][K = 2*(L/16)]
    v2f bm; bm.x = 1.0f; bm.y = 1.0f;    // B = ones(4x16)
    v8f cz = {};
    v8f dm = __builtin_amdgcn_wmma_f32_16x16x4_f32(
        /*neg_a=*/false, a, /*neg_b=*/false, bm,
        /*c_mod=*/(short)0, cz, /*reuse_a=*/false, /*reuse_b=*/false);
    float s = dm[0] + dm[1] + dm[2] + dm[3] + dm[4] + dm[5] + dm[6] + dm[7];
    s += __shfl_xor(s, 16, 32);          // combine lane-groups: full 32-lane sum

    __shared__ float wsum[8];
    const int lane = threadIdx.x & 31;
    const int wid  = threadIdx.x >> 5;
    if (lane == 0) wsum[wid] = s;
    __syncthreads();
    if (threadIdx.x == 0) {
        float bs = 0.0f;
#pragma unroll
        for (int i = 0; i < 8; ++i) bs += wsum[i];
        atomicAdd(out, bs * INV_N);      // global_atomic_add_f32, 1 per block
    }
}

extern "C" void kernel_launch(void* const* d_in, const int* in_sizes, int n_in,
                              void* d_out, int out_size, void* d_ws, size_t ws_size,
                              hipStream_t stream) {
    const float* pred  = (const float*)d_in[0];
    const float* truth = (const float*)d_in[1];
    const int*   mask  = (const int*)d_in[2];
    float* out = (float*)d_out;
    int*   ws  = (int*)d_ws;

    k_init<<<(WS_INTS + 255) / 256, 256, 0, stream>>>(ws, out);
    k_project<<<NB * SH, 256, 0, stream>>>(mask, ws, ws + NB * SH, ws + 2 * NB * SH);
    k_bounds<<<1, 32, 0, stream>>>(ws);
    k_mse<<<2048, 256, 0, stream>>>((const float4*)pred, (const float4*)truth, ws, out);
}